// GAT_12524124635295
// MI455X (gfx1250) — compile-verified
//
#include <hip/hip_runtime.h>
#include <math.h>

// ---------------------------------------------------------------------------
// GAT forward for MI455X (gfx1250, wave32, WMMA f32 16x16x4).
// N=4096, F_in=512, D=64, H=8.
// Round 3: 32-row waves (2 A fragments share B), software-pipelined loads,
// 32-bit offsets for saddr-form global loads. (Fixed col_cast decl order.)
// ---------------------------------------------------------------------------

typedef float v2f __attribute__((ext_vector_type(2)));
typedef float v8f __attribute__((ext_vector_type(8)));

#define GAT_N    4096
#define GAT_FIN  512
#define GAT_H    8
#define ALPHA    0.2f

static __device__ __forceinline__ v8f wmma4(v2f a, v2f b, v8f c) {
  // D = A(16x4 f32) * B(4x16 f32) + C(16x16 f32)
  return __builtin_amdgcn_wmma_f32_16x16x4_f32(
      false, a, false, b, (short)0, c, false, false);
}

static __device__ __forceinline__ float lrelu(float x) {
  return x > 0.0f ? x : ALPHA * x;
}
static __device__ __forceinline__ float elu1(float x) {
  return x > 0.0f ? x : (__expf(x) - 1.0f);
}

// ---------------------------------------------------------------------------
// Kernel 0: pack adjacency (int32 0/1, [N,N]) into bitmask [N, N/32].
// ---------------------------------------------------------------------------
__global__ void pack_adj_kernel(const int* __restrict__ adj,
                                unsigned* __restrict__ bits) {
  long long idx = (long long)blockIdx.x * blockDim.x + threadIdx.x;
  unsigned long long m = __ballot(adj[idx] != 0);
  if ((threadIdx.x & 31u) == 0u) bits[idx >> 5] = (unsigned)m;
}

// ---------------------------------------------------------------------------
// Kernel 1: f32 WMMA GEMM. One wave: 32 rows x 64 cols of
//   C[r, z*64 + c] = sum_k A[r,k] * B[z*K*ldb + k*ldb + c]
// grid.x = M/32, grid.z = #B-slices (heads). blockDim = 32.
// ---------------------------------------------------------------------------
__global__ void __launch_bounds__(32)
gemm_wmma_f32(const float* __restrict__ A, int lda,
              const float* __restrict__ B, int ldb,
              float* __restrict__ C, int ldc, int K) {
  const int lane = threadIdx.x;
  const int col  = lane & 15;
  const int hi   = lane >> 4;
  const int r0   = blockIdx.x * 32;
  const int z    = blockIdx.z;

  const float* Bz = B + (size_t)z * (size_t)K * (size_t)ldb;
  const unsigned aoff0 = (unsigned)(r0 + col) * (unsigned)lda;
  const unsigned aoff1 = aoff0 + 16u * (unsigned)lda;
  const unsigned uldb  = (unsigned)ldb;

  v8f ac00 = {}, ac01 = {}, ac02 = {}, ac03 = {};   // rows r0..r0+15
  v8f ac10 = {}, ac11 = {}, ac12 = {}, ac13 = {};   // rows r0+16..r0+31

#define GEMM_LOAD(kk, aa0, aa1, bv)                                          \
  do {                                                                       \
    const int ka_ = (kk) + 2 * hi;                                           \
    const unsigned bo_ = (unsigned)ka_ * uldb + (unsigned)col;               \
    bv[0] = Bz[bo_];           bv[1] = Bz[bo_ + uldb];                       \
    bv[2] = Bz[bo_ + 16];      bv[3] = Bz[bo_ + uldb + 16];                  \
    bv[4] = Bz[bo_ + 32];      bv[5] = Bz[bo_ + uldb + 32];                  \
    bv[6] = Bz[bo_ + 48];      bv[7] = Bz[bo_ + uldb + 48];                  \
    aa0.x = A[aoff0 + ka_];    aa0.y = A[aoff0 + ka_ + 1];                   \
    aa1.x = A[aoff1 + ka_];    aa1.y = A[aoff1 + ka_ + 1];                   \
  } while (0)

  float bb[8]; v2f a0, a1;
  GEMM_LOAD(0, a0, a1, bb);
  for (int k = 0; k < K; k += 4) {
    float nb[8]; v2f na0, na1;
    const int kn = (k + 4 < K) ? (k + 4) : 0;     // uniform; last iter dummy
    GEMM_LOAD(kn, na0, na1, nb);
    v2f bt;
    bt.x = bb[0]; bt.y = bb[1];
    ac00 = wmma4(a0, bt, ac00); ac10 = wmma4(a1, bt, ac10);
    bt.x = bb[2]; bt.y = bb[3];
    ac01 = wmma4(a0, bt, ac01); ac11 = wmma4(a1, bt, ac11);
    bt.x = bb[4]; bt.y = bb[5];
    ac02 = wmma4(a0, bt, ac02); ac12 = wmma4(a1, bt, ac12);
    bt.x = bb[6]; bt.y = bb[7];
    ac03 = wmma4(a0, bt, ac03); ac13 = wmma4(a1, bt, ac13);
    #pragma unroll
    for (int i = 0; i < 8; ++i) bb[i] = nb[i];
    a0 = na0; a1 = na1;
  }
#undef GEMM_LOAD

  float* Cp = C + (size_t)z * 64 + col;
  #pragma unroll
  for (int v = 0; v < 8; ++v) {
    const int rr = r0 + v + 8 * hi;               // D layout: row = vgpr+8*half
    float* cr0 = Cp + (size_t)rr * (size_t)ldc;
    float* cr1 = Cp + (size_t)(rr + 16) * (size_t)ldc;
    cr0[0] = ac00[v]; cr0[16] = ac01[v]; cr0[32] = ac02[v]; cr0[48] = ac03[v];
    cr1[0] = ac10[v]; cr1[16] = ac11[v]; cr1[32] = ac12[v]; cr1[48] = ac13[v];
  }
}

// ---------------------------------------------------------------------------
// Kernel 2: attention logit projections.
//   f_src[h*N+n] = dot(feat[n, h*64 : (h+1)*64], a_src[h]);  same for dst.
// ---------------------------------------------------------------------------
__global__ void dots_kernel(const float* __restrict__ feat, int ldf,
                            const float* __restrict__ a_src,
                            const float* __restrict__ a_dst,
                            float* __restrict__ f_src,
                            float* __restrict__ f_dst, int N) {
  const int idx = blockIdx.x * blockDim.x + threadIdx.x; // h*N + n
  const int h = idx / N;
  const int n = idx - h * N;
  const float* f  = feat + (size_t)n * (size_t)ldf + h * 64;
  const float* as = a_src + h * 64;
  const float* ad = a_dst + h * 64;
  float s = 0.0f, d = 0.0f;
  #pragma unroll 8
  for (int i = 0; i < 64; ++i) {
    const float v = f[i];
    s = fmaf(v, as[i], s);
    d = fmaf(v, ad[i], d);
  }
  f_src[idx] = s;
  f_dst[idx] = d;
}

// ---------------------------------------------------------------------------
// Kernel 3/6: fused masked-softmax attention + aggregation.
// One wave: (head = blockIdx.y, rows r0..r0+31), all 64 feature cols.
// Pass 1: masked max of fd per row (lrelu monotone => row max of e).
// Pass 2: p = exp(lrelu(fs+fd) - M) on edges; accumulate L + 8 WMMA tiles.
// Epilogue: scale by 1/L; FINAL=false -> ELU into h_cat[n, h*64+d];
//           FINAL=true  -> ELU + 64-wide log_softmax into out.
// ---------------------------------------------------------------------------
template <bool FINAL>
__global__ void __launch_bounds__(32)
gat_attn_kernel(const float* __restrict__ feat, int ldf,     // Wh values (B)
                const float* __restrict__ fsrc,
                const float* __restrict__ fdst,
                const unsigned* __restrict__ bits,           // [N, N/32]
                float* __restrict__ out, int ldo, int N) {
  const int lane = threadIdx.x;
  const int row  = lane & 15;
  const int hi   = lane >> 4;
  const int h    = blockIdx.y;
  const int r0   = blockIdx.x * 32;

  const float* fd_h = fdst + (size_t)h * (size_t)N;
  const float  fs0  = fsrc[(size_t)h * (size_t)N + (r0 + row)];
  const float  fs1  = fsrc[(size_t)h * (size_t)N + (r0 + 16 + row)];
  const unsigned* brow0 = bits + (size_t)(r0 + row)      * (size_t)(N >> 5);
  const unsigned* brow1 = bits + (size_t)(r0 + 16 + row) * (size_t)(N >> 5);

  // ---- Pass 1: masked max of fd over this half's column range ----
  float mx0 = -INFINITY, mx1 = -INFINITY;
  {
    const int cbeg = hi * (N >> 1);
    unsigned w0 = 0, w1 = 0;
    for (int c = cbeg; c < cbeg + (N >> 1); c += 4) {
      if ((c & 31) == 0) { w0 = brow0[c >> 5]; w1 = brow1[c >> 5]; }
      const float4 f4 = *(const float4*)(fd_h + c);
      const unsigned b0 = w0 >> (c & 31), b1 = w1 >> (c & 31);
      mx0 = fmaxf(mx0, (b0 & 1u) ? f4.x : -INFINITY);
      mx0 = fmaxf(mx0, (b0 & 2u) ? f4.y : -INFINITY);
      mx0 = fmaxf(mx0, (b0 & 4u) ? f4.z : -INFINITY);
      mx0 = fmaxf(mx0, (b0 & 8u) ? f4.w : -INFINITY);
      mx1 = fmaxf(mx1, (b1 & 1u) ? f4.x : -INFINITY);
      mx1 = fmaxf(mx1, (b1 & 2u) ? f4.y : -INFINITY);
      mx1 = fmaxf(mx1, (b1 & 4u) ? f4.z : -INFINITY);
      mx1 = fmaxf(mx1, (b1 & 8u) ? f4.w : -INFINITY);
    }
  }
  mx0 = fmaxf(mx0, __shfl_xor(mx0, 16));
  mx1 = fmaxf(mx1, __shfl_xor(mx1, 16));
  const float M0 = lrelu(fs0 + mx0);    // row max of masked e, rows r0..+15
  const float M1 = lrelu(fs1 + mx1);    // rows r0+16..+31

  // ---- Pass 2: exp + denominator + WMMA aggregation (pipelined B) ----
  v8f ac00 = {}, ac01 = {}, ac02 = {}, ac03 = {};
  v8f ac10 = {}, ac11 = {}, ac12 = {}, ac13 = {};
  float L0 = 0.0f, L1 = 0.0f;
  const unsigned colbase = (unsigned)h * 64u + (unsigned)row;
  const unsigned uldf = (unsigned)ldf;

#define ATTN_LOADB(mm, bv)                                                   \
  do {                                                                       \
    const unsigned bo_ = (unsigned)((mm) + 2 * hi) * uldf + colbase;         \
    bv[0] = feat[bo_];           bv[1] = feat[bo_ + uldf];                   \
    bv[2] = feat[bo_ + 16];      bv[3] = feat[bo_ + uldf + 16];              \
    bv[4] = feat[bo_ + 32];      bv[5] = feat[bo_ + uldf + 32];              \
    bv[6] = feat[bo_ + 48];      bv[7] = feat[bo_ + uldf + 48];              \
  } while (0)

  float bb[8];
  ATTN_LOADB(0, bb);
  unsigned w0 = 0, w1 = 0;
  for (int m = 0; m < N; m += 4) {
    float nb[8];
    const int mn = (m + 4 < N) ? (m + 4) : 0;     // uniform; last iter dummy
    ATTN_LOADB(mn, nb);
    if ((m & 31) == 0) { w0 = brow0[m >> 5]; w1 = brow1[m >> 5]; }
    const int c0 = m + 2 * hi;
    const float2 fd2 = *(const float2*)(fd_h + c0);
    const unsigned b0 = w0 >> (c0 & 31), b1 = w1 >> (c0 & 31);
    const float e00 = lrelu(fs0 + fd2.x), e01 = lrelu(fs0 + fd2.y);
    const float e10 = lrelu(fs1 + fd2.x), e11 = lrelu(fs1 + fd2.y);
    const float p00 = (b0 & 1u) ? __expf(e00 - M0) : 0.0f;
    const float p01 = (b0 & 2u) ? __expf(e01 - M0) : 0.0f;
    const float p10 = (b1 & 1u) ? __expf(e10 - M1) : 0.0f;
    const float p11 = (b1 & 2u) ? __expf(e11 - M1) : 0.0f;
    L0 += p00 + p01;
    L1 += p10 + p11;
    v2f a0; a0.x = p00; a0.y = p01;
    v2f a1; a1.x = p10; a1.y = p11;
    v2f bt;
    bt.x = bb[0]; bt.y = bb[1];
    ac00 = wmma4(a0, bt, ac00); ac10 = wmma4(a1, bt, ac10);
    bt.x = bb[2]; bt.y = bb[3];
    ac01 = wmma4(a0, bt, ac01); ac11 = wmma4(a1, bt, ac11);
    bt.x = bb[4]; bt.y = bb[5];
    ac02 = wmma4(a0, bt, ac02); ac12 = wmma4(a1, bt, ac12);
    bt.x = bb[6]; bt.y = bb[7];
    ac03 = wmma4(a0, bt, ac03); ac13 = wmma4(a1, bt, ac13);
    #pragma unroll
    for (int i = 0; i < 8; ++i) bb[i] = nb[i];
  }
#undef ATTN_LOADB

  L0 += __shfl_xor(L0, 16);
  L1 += __shfl_xor(L1, 16);
  const float invL0 = (L0 > 0.0f) ? (1.0f / L0) : 0.0f;
  const float invL1 = (L1 > 0.0f) ? (1.0f / L1) : 0.0f;

  float* op = out + (size_t)h * (FINAL ? 0 : 64) + row;
  #pragma unroll
  for (int v = 0; v < 8; ++v) {
    #pragma unroll
    for (int set = 0; set < 2; ++set) {
      const int rr = r0 + set * 16 + v + 8 * hi;
      const float s = __shfl(set ? invL1 : invL0, v + 8 * hi);
      float x0 = elu1((set ? ac10[v] : ac00[v]) * s);
      float x1 = elu1((set ? ac11[v] : ac01[v]) * s);
      float x2 = elu1((set ? ac12[v] : ac02[v]) * s);
      float x3 = elu1((set ? ac13[v] : ac03[v]) * s);
      float* o = op + (size_t)rr * (size_t)ldo;
      if (!FINAL) {
        o[0] = x0; o[16] = x1; o[32] = x2; o[48] = x3;
      } else {
        float mr = fmaxf(fmaxf(x0, x1), fmaxf(x2, x3));
        for (int msk = 1; msk < 16; msk <<= 1)
          mr = fmaxf(mr, __shfl_xor(mr, msk));
        float se = __expf(x0 - mr) + __expf(x1 - mr) +
                   __expf(x2 - mr) + __expf(x3 - mr);
        for (int msk = 1; msk < 16; msk <<= 1)
          se += __shfl_xor(se, msk);
        const float ls = mr + __logf(se);
        o[0] = x0 - ls; o[16] = x1 - ls; o[32] = x2 - ls; o[48] = x3 - ls;
      }
    }
  }
}

// ---------------------------------------------------------------------------
// Host-side orchestration
// ---------------------------------------------------------------------------
extern "C" void kernel_launch(void* const* d_in, const int* in_sizes, int n_in,
                              void* d_out, int out_size, void* d_ws, size_t ws_size,
                              hipStream_t stream) {
  (void)in_sizes; (void)n_in; (void)out_size; (void)ws_size;
  const float* x      = (const float*)d_in[0];  // [N, 512]
  const int*   adj    = (const int*)  d_in[1];  // [N, N]
  const float* W      = (const float*)d_in[2];  // [H, 512, 64]
  const float* a_src  = (const float*)d_in[3];  // [H, 64]
  const float* a_dst  = (const float*)d_in[4];  // [H, 64]
  const float* W_o    = (const float*)d_in[5];  // [512, 64]
  const float* ao_src = (const float*)d_in[6];  // [64]
  const float* ao_dst = (const float*)d_in[7];  // [64]
  float* outp = (float*)d_out;                  // [N, 64]

  const int N = GAT_N, FIN = GAT_FIN, H = GAT_H;

  size_t off = 0;
  auto take = [&](size_t bytes) -> void* {
    off = (off + 255) & ~(size_t)255;
    void* p = (char*)d_ws + off;
    off += bytes;
    return p;
  };
  unsigned* adj_bits = (unsigned*)take((size_t)N * (N / 32) * 4); // 2 MB
  float* Whcat = (float*)take((size_t)N * 512 * 4);               // 8 MB
  float* f_src = (float*)take((size_t)H * N * 4);
  float* f_dst = (float*)take((size_t)H * N * 4);
  float* h_cat = (float*)take((size_t)N * 512 * 4);               // 8 MB
  float* Who   = (float*)take((size_t)N * 64 * 4);                // 1 MB
  float* fo_src = (float*)take((size_t)N * 4);
  float* fo_dst = (float*)take((size_t)N * 4);

  // 0) adjacency -> bitmask
  pack_adj_kernel<<<(N * (long long)N) / 256, 256, 0, stream>>>(adj, adj_bits);

  // 1) Wh_cat[n, h*64+d] = x @ W   (z = head)
  gemm_wmma_f32<<<dim3(N / 32, 1, H), 32, 0, stream>>>(
      x, FIN, W, 64, Whcat, 512, FIN);

  // 2) f_src/f_dst [H,N]
  dots_kernel<<<(H * N) / 256, 256, 0, stream>>>(
      Whcat, 512, a_src, a_dst, f_src, f_dst, N);

  // 3) attention layer 1 (ELU epilogue) -> h_cat[n, h*64+d]
  gat_attn_kernel<false><<<dim3(N / 32, H), 32, 0, stream>>>(
      Whcat, 512, f_src, f_dst, adj_bits, h_cat, 512, N);

  // 4) Who = h_cat @ W_o   [N,64]
  gemm_wmma_f32<<<dim3(N / 32, 1, 1), 32, 0, stream>>>(
      h_cat, 512, W_o, 64, Who, 64, 512);

  // 5) fo_src/fo_dst [N]
  dots_kernel<<<N / 256, 256, 0, stream>>>(
      Who, 64, ao_src, ao_dst, fo_src, fo_dst, N);

  // 6) attention layer 2 + ELU + log_softmax -> out [N,64]
  gat_attn_kernel<true><<<dim3(N / 32, 1), 32, 0, stream>>>(
      Who, 64, fo_src, fo_dst, adj_bits, outp, 64, N);
}